// QuantumBranch_20564303413947
// MI455X (gfx1250) — compile-verified
//
#include <hip/hip_runtime.h>
#include <math.h>

#define N_QUBITS 8
#define PROJ_DIM 64
#define ITEMS_PER_BLOCK 16
#define LN_EPS 1e-5f

typedef __attribute__((ext_vector_type(2))) float v2f;
typedef __attribute__((ext_vector_type(8))) float v8f;

// Fast tanh for |x| small-ish (inputs ~N(0,1)); uses native v_exp_f32.
static __device__ __forceinline__ float fast_tanh(float x) {
  const float xc = fminf(fmaxf(x, -15.f), 15.f);
  const float e2 = __expf(2.f * xc);
  return 1.f - 2.f / (e2 + 1.f);
}

// Apply complex 2x2 gate U on state bit b.  Index of amplitude = lane*8 + r.
// Bits 0..2 live in-lane (r), bits 3..7 live across lanes.
static __device__ __forceinline__ void cgate(
    float re[8], float im[8], int b, int lane,
    float u00r, float u00i, float u01r, float u01i,
    float u10r, float u10i, float u11r, float u11i)
{
  if (b < 3) {
    const int bit = 1 << b;
#pragma unroll
    for (int r0 = 0; r0 < 8; ++r0) {
      if (r0 & bit) continue;
      const int r1 = r0 | bit;
      const float a0r = re[r0], a0i = im[r0];
      const float a1r = re[r1], a1i = im[r1];
      re[r0] = u00r*a0r - u00i*a0i + u01r*a1r - u01i*a1i;
      im[r0] = u00r*a0i + u00i*a0r + u01r*a1i + u01i*a1r;
      re[r1] = u10r*a0r - u10i*a0i + u11r*a1r - u11i*a1i;
      im[r1] = u10r*a0i + u10i*a0r + u11r*a1i + u11i*a1r;
    }
  } else {
    const int mask = 1 << (b - 3);
    const bool hi = (lane & mask) != 0;
    const float uar = hi ? u10r : u00r, uai = hi ? u10i : u00i;
    const float ubr = hi ? u11r : u01r, ubi = hi ? u11i : u01i;
#pragma unroll
    for (int r = 0; r < 8; ++r) {
      const float pr = __shfl_xor(re[r], mask, 32);
      const float pi = __shfl_xor(im[r], mask, 32);
      const float a0r = hi ? pr : re[r], a0i = hi ? pi : im[r];
      const float a1r = hi ? re[r] : pr, a1i = hi ? im[r] : pi;
      re[r] = uar*a0r - uai*a0i + ubr*a1r - ubi*a1i;
      im[r] = uar*a0i + uai*a0r + ubr*a1i + ubi*a1r;
    }
  }
}

// CNOT: new[i] = old[i ^ (ctrl(i) << bt)], ctrl from bit bc of i.
static __device__ __forceinline__ void cnot(
    float re[8], float im[8], int bc, int bt, int lane)
{
  if (bt < 3) {
    const int tbit = 1 << bt;
#pragma unroll
    for (int r0 = 0; r0 < 8; ++r0) {
      if (r0 & tbit) continue;
      const int r1 = r0 | tbit;
      const bool ctrl = (bc < 3) ? (((r0 >> bc) & 1) != 0)
                                 : (((lane >> (bc - 3)) & 1) != 0);
      const float s0r = re[r0], s0i = im[r0], s1r = re[r1], s1i = im[r1];
      re[r0] = ctrl ? s1r : s0r;  im[r0] = ctrl ? s1i : s0i;
      re[r1] = ctrl ? s0r : s1r;  im[r1] = ctrl ? s0i : s1i;
    }
  } else {
    const int mask = 1 << (bt - 3);
#pragma unroll
    for (int r = 0; r < 8; ++r) {
      const float pr = __shfl_xor(re[r], mask, 32);
      const float pi = __shfl_xor(im[r], mask, 32);
      const bool ctrl = (bc < 3) ? (((r >> bc) & 1) != 0)
                                 : (((lane >> (bc - 3)) & 1) != 0);
      re[r] = ctrl ? pr : re[r];
      im[r] = ctrl ? pi : im[r];
    }
  }
}

__global__ __launch_bounds__(512)
void quantum_branch_kernel(
    const float* __restrict__ x,      // (B, 8)
    const float* __restrict__ wts,    // (2, 8, 3)
    const float* __restrict__ W,      // (8, 64)
    const float* __restrict__ bias,   // (64)
    const float* __restrict__ gamma,  // (64)
    const float* __restrict__ beta,   // (64)
    float* __restrict__ out,          // (B, 64)
    int batch)
{
  __shared__ float q_lds[ITEMS_PER_BLOCK][N_QUBITS];
  __shared__ float h_lds[ITEMS_PER_BLOCK][PROJ_DIM];

  const int tid  = threadIdx.x;
  const int lane = tid & 31;
  const int wid  = tid >> 5;                              // wave = one batch row
  const int item = blockIdx.x * ITEMS_PER_BLOCK + wid;
  const int itemc = item < batch ? item : batch - 1;      // clamp for tail loads

  // ---------------- circuit: 256 amps = 8 complex per lane, in VGPRs ----
  float re[8], im[8];
#pragma unroll
  for (int r = 0; r < 8; ++r) { re[r] = 0.f; im[r] = 0.f; }
  re[0] = (lane == 0) ? 1.0f : 0.0f;                      // |0...0>

  const float PI = 3.14159265358979323846f;

  // batch-dependent RX(th) per qubit:  U = [[c, -i s],[-i s, c]]
  // th/2 in (-pi/2, pi/2): native v_sin/v_cos are exact enough here.
#pragma unroll
  for (int w = 0; w < 8; ++w) {
    const float a  = 0.5f * PI * fast_tanh(x[itemc * 8 + w]);
    const float ch = __cosf(a);
    const float sh = __sinf(a);
    cgate(re, im, 7 - w, lane,
          ch, 0.f,   0.f, -sh,
          0.f, -sh,  ch, 0.f);
  }

  // 2 layers: fixed Rot gates + CNOT ring (phases bounded in [0, 2pi))
#pragma unroll
  for (int l = 0; l < 2; ++l) {
#pragma unroll
    for (int w = 0; w < 8; ++w) {
      const float* p = wts + (l * 8 + w) * 3;
      const float phi = p[0], theta = p[1], omega = p[2];
      const float st = __sinf(0.5f * theta), ct = __cosf(0.5f * theta);
      const float aa = 0.5f * (phi + omega), dd = 0.5f * (phi - omega);
      const float sa = __sinf(aa), ca = __cosf(aa);
      const float sd = __sinf(dd), cd = __cosf(dd);
      // m00 = e^{-ia} ct, m01 = -e^{id} st, m10 = e^{-id} st, m11 = e^{ia} ct
      cgate(re, im, 7 - w, lane,
            ct * ca, -ct * sa,
           -st * cd, -st * sd,
            st * cd, -st * sd,
            ct * ca,  ct * sa);
    }
    const int rr = (l % 7) + 1;
#pragma unroll
    for (int i = 0; i < 8; ++i) {
      const int c = i, t = (i + rr) & 7;
      cnot(re, im, 7 - c, 7 - t, lane);
    }
  }

  // ---------------- Z expectations: z_w = sum p_i * (1 - 2*bit_{7-w}(i)) ----
  float z[8];
#pragma unroll
  for (int w = 0; w < 8; ++w) z[w] = 0.f;
#pragma unroll
  for (int r = 0; r < 8; ++r) {
    const float p = re[r] * re[r] + im[r] * im[r];
    const int idx = lane * 8 + r;
#pragma unroll
    for (int w = 0; w < 8; ++w)
      z[w] += ((idx >> (7 - w)) & 1) ? -p : p;
  }
#pragma unroll
  for (int off = 16; off >= 1; off >>= 1) {
#pragma unroll
    for (int w = 0; w < 8; ++w)
      z[w] += __shfl_xor(z[w], off, 32);
  }
  if (lane == 0) {
#pragma unroll
    for (int w = 0; w < 8; ++w) q_lds[wid][w] = z[w];
  }
  __syncthreads();

  // ---------------- projection via WMMA f32 16x16x4 (K=8 -> 2 WMMAs) ----
  // waves 0..3 each own one 16-wide N tile of the (16 x 64) output block.
  if (wid < 4) {
    const int half = lane >> 4;          // 0: K pair {0,1}/{4,5}; 1: {2,3}/{6,7}
    const int lm   = lane & 15;
    const int n    = wid * 16 + lm;

    v2f a0, a1, b0, b1;
    // A (16x4) layout: lanes 0-15 -> K0,K1 ; lanes 16-31 -> K2,K3
    a0[0] = q_lds[lm][half * 2 + 0];
    a0[1] = q_lds[lm][half * 2 + 1];
    a1[0] = q_lds[lm][half * 2 + 4];
    a1[1] = q_lds[lm][half * 2 + 5];
    // B (4x16): row k striped across lanes, K split like A
    b0[0] = W[(half * 2 + 0) * PROJ_DIM + n];
    b0[1] = W[(half * 2 + 1) * PROJ_DIM + n];
    b1[0] = W[(half * 2 + 4) * PROJ_DIM + n];
    b1[1] = W[(half * 2 + 5) * PROJ_DIM + n];

    v8f c = {};
    c = __builtin_amdgcn_wmma_f32_16x16x4_f32(false, a0, false, b0,
                                              (short)0, c, false, false);
    c = __builtin_amdgcn_wmma_f32_16x16x4_f32(false, a1, false, b1,
                                              (short)0, c, false, false);

    const float bb = bias[n];
    // C/D layout: VGPR r -> M=r (lanes 0-15) / M=r+8 (lanes 16-31), N=lane&15
#pragma unroll
    for (int r = 0; r < 8; ++r)
      h_lds[r + half * 8][n] = c[r] + bb;
  }
  __syncthreads();

  // ---------------- LayerNorm: wave `wid` normalizes row `wid` ----
  const float h0 = h_lds[wid][lane];
  const float h1 = h_lds[wid][lane + 32];
  float s  = h0 + h1;
  float s2 = h0 * h0 + h1 * h1;
#pragma unroll
  for (int off = 16; off >= 1; off >>= 1) {
    s  += __shfl_xor(s,  off, 32);
    s2 += __shfl_xor(s2, off, 32);
  }
  const float mu   = s * (1.0f / 64.0f);
  const float var  = s2 * (1.0f / 64.0f) - mu * mu;
  const float rinv = rsqrtf(var + LN_EPS);

  if (item < batch) {
    out[(size_t)item * 64 + lane]      = gamma[lane]      * (h0 - mu) * rinv + beta[lane];
    out[(size_t)item * 64 + lane + 32] = gamma[lane + 32] * (h1 - mu) * rinv + beta[lane + 32];
  }
}

extern "C" void kernel_launch(void* const* d_in, const int* in_sizes, int n_in,
                              void* d_out, int out_size, void* d_ws, size_t ws_size,
                              hipStream_t stream) {
  const float* x     = (const float*)d_in[0];
  const float* wts   = (const float*)d_in[1];
  const float* W     = (const float*)d_in[2];
  const float* bias  = (const float*)d_in[3];
  const float* gamma = (const float*)d_in[4];
  const float* beta  = (const float*)d_in[5];
  float* out = (float*)d_out;

  const int batch  = in_sizes[0] / N_QUBITS;
  const int blocks = (batch + ITEMS_PER_BLOCK - 1) / ITEMS_PER_BLOCK;

  quantum_branch_kernel<<<blocks, 512, 0, stream>>>(x, wts, W, bias, gamma, beta,
                                                    out, batch);
}